// Encoder_24154896072769
// MI455X (gfx1250) — compile-verified
//
#include <hip/hip_runtime.h>
#include <hip/hip_bf16.h>
#include <stdint.h>

#define B_DIM 32
#define T_TILE 256

// ---------------------------------------------------------------------------
// Kernel 1: transpose x (B, N_IN) -> xT (N_IN, B) using CDNA5 async
// global->LDS loads in GVS addressing mode:
//   mem_addr = SGPR64(base for batch b) + VGPR32(i*4)
// The async op takes a per-lane LDS destination address, so the transpose
// scatter happens inside the LDS write (no VGPR round trip). LDS row pitch
// 33 floats -> bank-conflict-free for both scatter and read.
// ---------------------------------------------------------------------------
__global__ __launch_bounds__(256) void xpose_async_kernel(
    const float* __restrict__ x, float* __restrict__ xT, int n_in) {
  __shared__ float tile[T_TILE * 33];  // 33,792 B
  const int tid = threadIdx.x;
  const int base = blockIdx.x * T_TILE;
  const int i = base + tid;

  if (i < n_in) {
    // low 32 bits of a generic pointer to LDS == LDS byte offset
    const unsigned lds  = (unsigned)(uintptr_t)&tile[tid * 33];
    const unsigned voff = (unsigned)i * 4u;  // per-lane byte offset (coalesced)
#pragma unroll
    for (int b = 0; b < B_DIM; ++b) {
      const float* sbase = x + (size_t)b * (size_t)n_in;  // wave-uniform -> SGPR pair
      // 128B coalesced global read per wave, scattered LDS write (transpose)
      asm volatile("global_load_async_to_lds_b32 %0, %1, %2"
                   :: "v"(lds + 4u * (unsigned)b), "v"(voff), "s"(sbase)
                   : "memory");
    }
  }
  asm volatile("s_wait_asynccnt 0" ::: "memory");
  __syncthreads();

  const int lane = tid & 31;   // = b
  const int wid  = tid >> 5;   // 0..7
#pragma unroll
  for (int r = 0; r < T_TILE / 8; ++r) {
    const int row  = wid + r * 8;
    const int irow = base + row;
    if (irow < n_in)
      xT[(size_t)irow * B_DIM + lane] = tile[row * 33 + lane];  // 128B store
  }
}

// ---------------------------------------------------------------------------
// Kernel 2: gather + weighted sum. lane == batch (wave32 == B == 32).
// Each wave owns a contiguous range of outputs. idx/w reads are wave-uniform
// (readfirstlane -> SMEM scalar loads). Each xT gather is one contiguous
// 128B line (xT is L2-resident). Results staged in LDS per 32-output chunk
// so every global store of `out` is a full coalesced 128B line.
// ---------------------------------------------------------------------------
template <bool COAL>
__global__ __launch_bounds__(256) void gather_dot_kernel(
    const float* __restrict__ xsrc,  // COAL: xT (N_IN,32); else x (32,N_IN)
    const float* __restrict__ w,
    const int*   __restrict__ idx,
    float*       __restrict__ out,
    int n_in, int n_out, int nwaves_total) {
  __shared__ float stage[8 * 32 * 33];  // 33,792 B
  const int lane = threadIdx.x & 31;    // batch index b
  const int wid  = threadIdx.x >> 5;
  float* wtile = &stage[wid * 32 * 33];

  const int gwave   = blockIdx.x * 8 + wid;
  const int per     = (n_out + nwaves_total - 1) / nwaves_total;
  const int o_begin = gwave * per;
  int o_end = o_begin + per;
  if (o_end > n_out) o_end = n_out;

  for (int o0 = o_begin; o0 < o_end; o0 += 32) {
    const int jmax = (o_end - o0 < 32) ? (o_end - o0) : 32;
#pragma unroll 4
    for (int j = 0; j < jmax; ++j) {
      const int o  = __builtin_amdgcn_readfirstlane(o0 + j);  // force SMEM
      const int i0 = idx[3 * o + 0];
      const int i1 = idx[3 * o + 1];
      const int i2 = idx[3 * o + 2];
      const float w0 = w[3 * o + 0];
      const float w1 = w[3 * o + 1];
      const float w2 = w[3 * o + 2];
      float g0, g1, g2;
      if (COAL) {
        g0 = xsrc[(size_t)i0 * B_DIM + lane];   // contiguous 128B per wave
        g1 = xsrc[(size_t)i1 * B_DIM + lane];
        g2 = xsrc[(size_t)i2 * B_DIM + lane];
      } else {
        const size_t rb = (size_t)lane * (size_t)n_in;
        g0 = xsrc[rb + i0];
        g1 = xsrc[rb + i1];
        g2 = xsrc[rb + i2];
      }
      wtile[j * 33 + lane] = fmaf(g0, w0, fmaf(g1, w1, g2 * w2));
    }
    // write-out: value for (b, o0+lane) lives at wtile[lane*33 + b]
    if (o0 + lane < o_end) {
#pragma unroll
      for (int b = 0; b < B_DIM; ++b) {
        out[(size_t)b * (size_t)n_out + (size_t)(o0 + lane)] =
            wtile[lane * 33 + b];  // full 128B line per wave store
      }
    }
  }
}

// ---------------------------------------------------------------------------
extern "C" void kernel_launch(void* const* d_in, const int* in_sizes, int n_in_args,
                              void* d_out, int out_size, void* d_ws, size_t ws_size,
                              hipStream_t stream) {
  (void)n_in_args; (void)out_size;
  const float* x   = (const float*)d_in[0];
  const float* w   = (const float*)d_in[1];
  const int*   idx = (const int*)d_in[2];
  float*       out = (float*)d_out;

  const int n_in  = in_sizes[0] / B_DIM;  // 1,000,000
  const int n_out = in_sizes[2] / 3;      // 500,000

  const int gblocks = 1024;               // 8192 waves -> ~62 outputs each
  const int nwaves  = gblocks * 8;

  const size_t need = (size_t)n_in * B_DIM * sizeof(float);  // 128 MB
  if (ws_size >= need) {
    float* xT = (float*)d_ws;
    const int tblocks = (n_in + T_TILE - 1) / T_TILE;
    hipLaunchKernelGGL(xpose_async_kernel, dim3(tblocks), dim3(256), 0, stream,
                       x, xT, n_in);
    hipLaunchKernelGGL((gather_dot_kernel<true>), dim3(gblocks), dim3(256), 0,
                       stream, xT, w, idx, out, n_in, n_out, nwaves);
  } else {
    hipLaunchKernelGGL((gather_dot_kernel<false>), dim3(gblocks), dim3(256), 0,
                       stream, x, w, idx, out, n_in, n_out, nwaves);
  }
}